// NonMaximaSuppression2d_49529562858245
// MI455X (gfx1250) — compile-verified
//
#include <hip/hip_runtime.h>
#include <stdint.h>

// NMS2d: out = x * (x > max(0, 8 replicate-padded 3x3 neighbors))
// Memory-bound: 512 MB traffic @ 23.3 TB/s -> ~22us floor. Strategy:
// LDS-tiled stencil, halo strip copied with CDNA5 async-to-LDS bulk loads
// (ASYNCcnt), sliding 3-row register window over LDS, NT b128 stores.

#define HH 512
#define WW 512
#define TILE_H 16
#define NTHREADS 256

typedef __attribute__((ext_vector_type(4))) float float4v;

__global__ __launch_bounds__(NTHREADS) void nms2d_kernel(const float* __restrict__ x,
                                                         float* __restrict__ out) {
  __shared__ float lds[TILE_H + 2][WW];

  const int tid     = threadIdx.x;
  const int img     = blockIdx.x >> 5;   // 32 row-tiles per 512-row image
  const int tileY   = blockIdx.x & 31;
  const int rowBase = tileY * TILE_H;
  const int half    = tid >> 7;          // 0 or 1
  const int lane16  = tid & 127;         // 128 lanes x b128 = one 512-float row

  const uint64_t base = (uint64_t)(uintptr_t)(x + (size_t)img * HH * WW);

  // ---- async halo-strip copy: (TILE_H+2) rows x 512 floats into LDS ----
  // Uniform trip count -> straight-line unrolled async loads, no exec games.
#pragma unroll
  for (int it = 0; it < (TILE_H + 2) / 2; ++it) {
    const int i = it * 2 + half;                        // 0..17
    int srcRow = rowBase - 1 + i;                       // replicate-pad in H
    srcRow = srcRow < 0 ? 0 : (srcRow > HH - 1 ? HH - 1 : srcRow);
    const uint32_t goff = (uint32_t)srcRow * (WW * 4) + (uint32_t)lane16 * 16;
    const uint32_t loff = (uint32_t)(uintptr_t)(&lds[i][lane16 * 4]);
    asm volatile("global_load_async_to_lds_b128 %0, %1, %2"
                 :
                 : "v"(loff), "v"(goff), "s"(base)
                 : "memory");
  }
  asm volatile("s_wait_asynccnt 0" ::: "memory");
  __syncthreads();

  // ---- compute: thread-half owns 8 contiguous rows; 3-row sliding window ----
  float* dst = out + (size_t)img * HH * WW;
  const int c0 = lane16 * 4;
  const int cL = (c0 == 0) ? 0 : c0 - 1;                // replicate-pad in W
  const int cR = (c0 + 4 > WW - 1) ? WW - 1 : c0 + 4;

  float vals[3][6];
#define LOAD_ROW(ldsRow, dstRow)                                    \
  {                                                                 \
    const float* row_ = &lds[(ldsRow)][0];                          \
    const float4v v_ = *(const float4v*)(row_ + c0);                \
    vals[(dstRow)][0] = row_[cL];                                   \
    vals[(dstRow)][1] = v_.x;                                       \
    vals[(dstRow)][2] = v_.y;                                       \
    vals[(dstRow)][3] = v_.z;                                       \
    vals[(dstRow)][4] = v_.w;                                       \
    vals[(dstRow)][5] = row_[cR];                                   \
  }

  const int r0 = half * (TILE_H / 2);                   // 0 or 8
  LOAD_ROW(r0 + 0, 0)                                   // lds row (r-1)+1
  LOAD_ROW(r0 + 1, 1)

#pragma unroll
  for (int it = 0; it < TILE_H / 2; ++it) {
    const int r = r0 + it;
    const int top = it % 3, mid = (it + 1) % 3, bot = (it + 2) % 3;
    LOAD_ROW(r + 2, bot)                                // 1 new row per output row

    float4v o;
#pragma unroll
    for (int j = 0; j < 4; ++j) {
      float m = 0.0f;                                   // zeroed center tap
      m = fmaxf(m, vals[top][j + 0]);
      m = fmaxf(m, vals[top][j + 1]);
      m = fmaxf(m, vals[top][j + 2]);
      m = fmaxf(m, vals[mid][j + 0]);
      m = fmaxf(m, vals[mid][j + 2]);                   // skip center
      m = fmaxf(m, vals[bot][j + 0]);
      m = fmaxf(m, vals[bot][j + 1]);
      m = fmaxf(m, vals[bot][j + 2]);
      const float c = vals[mid][j + 1];
      ((float*)&o)[j] = (c > m) ? c : 0.0f;
    }
    __builtin_nontemporal_store(o, (float4v*)(dst + (size_t)(rowBase + r) * WW + c0));
  }
#undef LOAD_ROW
}

extern "C" void kernel_launch(void* const* d_in, const int* in_sizes, int n_in,
                              void* d_out, int out_size, void* d_ws, size_t ws_size,
                              hipStream_t stream) {
  (void)n_in; (void)d_ws; (void)ws_size; (void)out_size;
  const float* x = (const float*)d_in[0];
  float* out = (float*)d_out;
  const int n_img = in_sizes[0] / (HH * WW);            // B*C = 256 images
  dim3 grid(n_img * (HH / TILE_H));                     // 8192 blocks
  nms2d_kernel<<<grid, NTHREADS, 0, stream>>>(x, out);
}